// DAWNBlock_25864293056822
// MI455X (gfx1250) — compile-verified
//
#include <hip/hip_runtime.h>
#include <hip/hip_bf16.h>
#include <math.h>

// ---------------------------------------------------------------------------
// DAWN block for MI455X (gfx1250, wave32).
// Stage 1: h = x @ W_proj + b   via V_WMMA_F32_16X16X4_F32, with double-
//          buffered GLOBAL_LOAD_ASYNC_TO_LDS_B128 streaming (ASYNCcnt) when
//          the toolchain exposes the builtin.
// Stage 2: logits = h @ emb_n^T, fused softmax (no-max form) + importance
//          pooling, accumulated to dense[4][1536] with global f32 atomics.
// Stage 3: per-row top-k + renormalize (tiny).
// ---------------------------------------------------------------------------

typedef __attribute__((ext_vector_type(2))) float v2f;
typedef __attribute__((ext_vector_type(4))) float v4f;
typedef __attribute__((ext_vector_type(8))) float v8f;

// exact param type of the async-LDS builtin (from hipcc diagnostic):
// 'int __attribute__((vector_size(4*sizeof(int))))' with AS qualifiers
typedef int v4i_vs __attribute__((vector_size(4 * sizeof(int))));
typedef __attribute__((address_space(1))) v4i_vs* g_v4i_p;
typedef __attribute__((address_space(3))) v4i_vs* l_v4i_p;

#define B_    4
#define S_    4096
#define D_    2048
#define DS_   64
#define NTOT  1536
#define NG    512
#define ROWS  64     // tokens per block
#define KC    64     // k-chunk for stage-1 streaming
#define XSTR  68     // padded LDS row stride (floats): 68*4 = 272B = 17*16
#define EXSTR 513    // padded stride for exp-logits buffer (bank spread)

#if __has_builtin(__builtin_amdgcn_global_load_async_to_lds_b128)
#define HAVE_ASYNC_LDS 1
#else
#define HAVE_ASYNC_LDS 0
#endif

__device__ inline v8f zero8() {
    v8f z;
#pragma unroll
    for (int i = 0; i < 8; ++i) z[i] = 0.f;
    return z;
}

__device__ inline void wait_async0() {
#if __has_builtin(__builtin_amdgcn_s_wait_asynccnt)
    __builtin_amdgcn_s_wait_asynccnt(0);
#else
    asm volatile("s_wait_asynccnt 0x0" ::: "memory");
#endif
}

#if HAVE_ASYNC_LDS
// 16B per-lane async copy: global -> LDS, tracked by ASYNCcnt.
__device__ inline void async_copy16(const float* g, float* l) {
    __builtin_amdgcn_global_load_async_to_lds_b128(
        (g_v4i_p)(g), (l_v4i_p)(l), 0, 0);
}
#endif

// --------------------------- prep: normalize emb rows, zero dense ----------
__global__ void dawn_prep(const float* __restrict__ emb,
                          float* __restrict__ embn,
                          float* __restrict__ dense) {
    int idx = blockIdx.x * blockDim.x + threadIdx.x;
    if (idx < B_ * NTOT) dense[idx] = 0.f;
    if (idx < NTOT) {
        const float* r = emb + (size_t)idx * DS_;
        float s = 0.f;
#pragma unroll 8
        for (int i = 0; i < DS_; ++i) { float v = r[i]; s += v * v; }
        float inv = 1.f / fmaxf(sqrtf(s), 1e-12f);
        float* o = embn + (size_t)idx * DS_;
#pragma unroll 8
        for (int i = 0; i < DS_; ++i) o[i] = r[i] * inv;
    }
}

// --------------------------- main fused kernel -----------------------------
__global__ __launch_bounds__(256) void dawn_main(
    const float* __restrict__ x, const float* __restrict__ imp_g,
    const float* __restrict__ Wp, const float* __restrict__ bp,
    const float* __restrict__ embn, float* __restrict__ dense) {
    extern __shared__ float smem[];
    float* hb   = smem;                    // 64*68  h = x@W+b (persists)
    float* impL = hb + ROWS * XSTR;        // 64
    float* rsum = impL + ROWS;             // 64
    float* fL   = rsum + ROWS;             // 64
    float* big  = fL + ROWS;               // max(4 stage-1 bufs, ex) = 64*513
    float* Xb0  = big;                     // 64*68
    float* Wb0  = big + ROWS * XSTR;       // 64*68
#if HAVE_ASYNC_LDS
    float* Xb1  = big + 2 * ROWS * XSTR;   // 64*68 (double buffer)
    float* Wb1  = big + 3 * ROWS * XSTR;   // 64*68
#endif
    float* ex   = big;                     // 64*513 (stage-2 only)

    const int tid    = threadIdx.x;
    const int w      = tid >> 5;          // wave 0..7
    const int lane   = tid & 31;
    const int laneN  = lane & 15;
    const int hiHalf = (lane >> 4) & 1;   // lanes 16-31
    const int row_base = blockIdx.x * ROWS;
    const int b = row_base / S_;

    // wave w owns stage-1 tiles (m0, n0) and (m0+2, n0)
    const int m0 = (w >> 2);
    const int n0 = (w & 3);

    v8f acc0 = zero8(), acc1 = zero8();

    // WMMA over one staged K-chunk
    auto compute_chunk = [&](const float* Xb, const float* Wb) {
#pragma unroll
        for (int kk = 0; kk < KC; kk += 4) {
            const int ko = kk + 2 * hiHalf;
            // B frag (4x16 of W): shared by both m-tiles of this wave
            v2f bfr;
            bfr[0] = Wb[(ko)     * XSTR + n0 * 16 + laneN];
            bfr[1] = Wb[(ko + 1) * XSTR + n0 * 16 + laneN];
            // A frags (16x4 of x): one b64 LDS load per lane
            v2f a0 = *(const v2f*)(Xb + (m0 * 16 + laneN) * XSTR + ko);
            v2f a1 = *(const v2f*)(Xb + ((m0 + 2) * 16 + laneN) * XSTR + ko);
            acc0 = __builtin_amdgcn_wmma_f32_16x16x4_f32(
                false, a0, false, bfr, (short)0, acc0, false, false);
            acc1 = __builtin_amdgcn_wmma_f32_16x16x4_f32(
                false, a1, false, bfr, (short)0, acc1, false, false);
        }
    };

    // -------- stage 1: h = x @ W, K streamed in 64-wide chunks -------------
#if HAVE_ASYNC_LDS
    // Async copy-engine path: double-buffered, copy of chunk kc+1 overlaps
    // WMMA on chunk kc. One s_wait_asynccnt + barrier per chunk.
    auto issue_chunk = [&](int k0, float* Xb, float* Wb) {
#pragma unroll
        for (int j = 0; j < 4; ++j) {
            int idx4 = tid + j * 256;           // 1024 16B slots per matrix
            int r  = idx4 >> 4;
            int c4 = (idx4 & 15) << 2;
            async_copy16(x + (size_t)(row_base + r) * D_ + k0 + c4,
                         Xb + r * XSTR + c4);
            async_copy16(Wp + (size_t)(k0 + r) * DS_ + c4,
                         Wb + r * XSTR + c4);
        }
    };

    issue_chunk(0, Xb0, Wb0);
    wait_async0();
    __syncthreads();
    for (int kc = 0; kc < D_ / KC; ++kc) {
        float* Xc = (kc & 1) ? Xb1 : Xb0;
        float* Wc = (kc & 1) ? Wb1 : Wb0;
        if (kc + 1 < D_ / KC)
            issue_chunk((kc + 1) * KC, (kc & 1) ? Xb0 : Xb1,
                                       (kc & 1) ? Wb0 : Wb1);
        compute_chunk(Xc, Wc);
        wait_async0();      // next chunk landed in LDS
        __syncthreads();    // everyone done reading current chunk
    }
#else
    // Fallback: VGPR-staged loads, single buffer.
    for (int k0 = 0; k0 < D_; k0 += KC) {
        __syncthreads();   // previous chunk fully consumed
#pragma unroll
        for (int j = 0; j < 4; ++j) {
            int idx4 = tid + j * 256;
            int r  = idx4 >> 4;
            int c4 = (idx4 & 15) << 2;
            v4f xv = *(const v4f*)(x + (size_t)(row_base + r) * D_ + k0 + c4);
            *(v4f*)(Xb0 + r * XSTR + c4) = xv;
            v4f wv = *(const v4f*)(Wp + (size_t)(k0 + r) * DS_ + c4);
            *(v4f*)(Wb0 + r * XSTR + c4) = wv;
        }
        if (k0 + KC < D_)
            __builtin_prefetch(x + (size_t)(row_base + (tid & 63)) * D_ + k0 + KC, 0, 1);
        __syncthreads();
        compute_chunk(Xb0, Wb0);
    }
    __syncthreads();
#endif

    // bias add + park h in LDS (C/D layout: VGPR r -> M = r + 8*hiHalf)
    const float biasv = bp[n0 * 16 + laneN];
#pragma unroll
    for (int r = 0; r < 8; ++r) {
        int row0 = m0 * 16 + r + 8 * hiHalf;
        int row1 = (m0 + 2) * 16 + r + 8 * hiHalf;
        hb[row0 * XSTR + n0 * 16 + laneN] = acc0[r] + biasv;
        hb[row1 * XSTR + n0 * 16 + laneN] = acc1[r] + biasv;
    }
    if (tid < ROWS) impL[tid] = imp_g[(size_t)b * S_ + row_base + tid];
    __syncthreads();   // h + imp visible; stage-1 staging region now free

    // -------- stage 2: per group, logits -> exp -> pooled dense ------------
    for (int g = 0; g < 3; ++g) {
#pragma unroll 1
        for (int nt = 0; nt < 4; ++nt) {
            const int n_tile = w * 4 + nt;            // 0..31
            const int ncol   = n_tile * 16 + laneN;   // 0..511
            const float* embp = embn + (size_t)(g * NG + ncol) * DS_;
            v8f c0 = zero8(), c1 = zero8(), c2 = zero8(), c3 = zero8();
#pragma unroll
            for (int kk = 0; kk < DS_; kk += 4) {
                const int ko = kk + 2 * hiHalf;
                v2f bfr = *(const v2f*)(embp + ko);   // normalized emb (L2-hot)
                v2f a0 = *(const v2f*)(hb + (0 * 16 + laneN) * XSTR + ko);
                v2f a1 = *(const v2f*)(hb + (1 * 16 + laneN) * XSTR + ko);
                v2f a2 = *(const v2f*)(hb + (2 * 16 + laneN) * XSTR + ko);
                v2f a3 = *(const v2f*)(hb + (3 * 16 + laneN) * XSTR + ko);
                c0 = __builtin_amdgcn_wmma_f32_16x16x4_f32(false, a0, false, bfr, (short)0, c0, false, false);
                c1 = __builtin_amdgcn_wmma_f32_16x16x4_f32(false, a1, false, bfr, (short)0, c1, false, false);
                c2 = __builtin_amdgcn_wmma_f32_16x16x4_f32(false, a2, false, bfr, (short)0, c2, false, false);
                c3 = __builtin_amdgcn_wmma_f32_16x16x4_f32(false, a3, false, bfr, (short)0, c3, false, false);
            }
            // exp() and park in padded LDS buffer
#pragma unroll
            for (int r = 0; r < 8; ++r) {
                int rr = r + 8 * hiHalf;
                int cc = n_tile * 16 + laneN;
                ex[(rr)      * EXSTR + cc] = __expf(c0[r]);
                ex[(16 + rr) * EXSTR + cc] = __expf(c1[r]);
                ex[(32 + rr) * EXSTR + cc] = __expf(c2[r]);
                ex[(48 + rr) * EXSTR + cc] = __expf(c3[r]);
            }
        }
        __syncthreads();

        // per-row softmax denominator (no-max form; |logit| <~ 12 so safe)
        if (tid < ROWS) {
            float s = 0.f;
            for (int c = 0; c < NG; ++c) s += ex[tid * EXSTR + c];
            rsum[tid] = s;
            fL[tid] = impL[tid] / s;   // importance / denom
        }
        __syncthreads();

        // pooled column sums -> global accumulator
        for (int c = tid; c < NG; c += 256) {
            float s = 0.f;
#pragma unroll 4
            for (int m = 0; m < ROWS; ++m) s += ex[m * EXSTR + c] * fL[m];
            atomicAdd(dense + (size_t)b * NTOT + g * NG + c, s);
        }
        __syncthreads();   // ex reused next group
    }
}

// --------------------------- top-k sparsify + renormalize ------------------
__global__ __launch_bounds__(256) void dawn_topk(const float* __restrict__ dense,
                                                 float* __restrict__ out) {
    __shared__ float v[NG];
    __shared__ float wk[NG];
    __shared__ int   sel[NG];
    __shared__ float rv[256];
    __shared__ int   ri[256];
    __shared__ float ssum;

    const int t    = threadIdx.x;
    const int slot = blockIdx.x >> 2;   // 0:C 1:Q 2:K 3:V
    const int b    = blockIdx.x & 3;
    const int g    = (slot == 0) ? 0 : (slot == 3 ? 2 : 1);
    const int k    = (slot == 0) ? 8 : (slot == 3 ? 6 : 4);
    const float* src = dense + (size_t)b * NTOT + g * NG;

    for (int c = t; c < NG; c += 256) {
        float xv = src[c];
        v[c] = xv; wk[c] = xv; sel[c] = 0;
    }
    if (t == 0) ssum = 0.f;
    __syncthreads();

    for (int it = 0; it < k; ++it) {
        float m = wk[t]; int mi = t;
        float m2 = wk[t + 256];
        if (m2 > m) { m = m2; mi = t + 256; }
        rv[t] = m; ri[t] = mi;
        __syncthreads();
        for (int sft = 128; sft > 0; sft >>= 1) {
            if (t < sft) {
                float ov = rv[t + sft]; int oi = ri[t + sft];
                if (ov > rv[t] || (ov == rv[t] && oi < ri[t])) { rv[t] = ov; ri[t] = oi; }
            }
            __syncthreads();
        }
        if (t == 0) { int ix = ri[0]; sel[ix] = 1; ssum += rv[0]; wk[ix] = -1e30f; }
        __syncthreads();
    }

    const float inv = 1.f / (ssum + 1e-8f);
    for (int c = t; c < NG; c += 256)
        out[((size_t)slot * B_ + b) * NG + c] = sel[c] ? v[c] * inv : 0.f;
}

// --------------------------- launch ----------------------------------------
extern "C" void kernel_launch(void* const* d_in, const int* in_sizes, int n_in,
                              void* d_out, int out_size, void* d_ws, size_t ws_size,
                              hipStream_t stream) {
    const float* x   = (const float*)d_in[0];
    const float* imp = (const float*)d_in[1];
    const float* Wp  = (const float*)d_in[2];
    const float* bp  = (const float*)d_in[3];
    const float* emb = (const float*)d_in[4];
    float* out = (float*)d_out;

    float* embn  = (float*)d_ws;                  // 1536*64 f32
    float* dense = embn + (size_t)NTOT * DS_;     // 4*1536 f32

    dawn_prep<<<(B_ * NTOT + 255) / 256, 256, 0, stream>>>(emb, embn, dense);

    const size_t shmem =
        (size_t)(ROWS * XSTR + 3 * ROWS + ROWS * EXSTR) * sizeof(float); // ~146 KB
    dawn_main<<<(B_ * S_) / ROWS, 256, shmem, stream>>>(x, imp, Wp, bp, embn, dense);

    dawn_topk<<<16, 256, 0, stream>>>(dense, out);
}